// MultivariateGaussianLoss_30794915512936
// MI455X (gfx1250) — compile-verified
//
#include <hip/hip_runtime.h>
#include <math.h>

#define AS1 __attribute__((address_space(1)))
#define AS3 __attribute__((address_space(3)))

typedef int v4i __attribute__((ext_vector_type(4)));

#ifndef __has_builtin
#define __has_builtin(x) 0
#endif

// ---- CDNA5 async global->LDS copy (ASYNCcnt-tracked, 16B per lane) ----
__device__ __forceinline__ void async_g2l_b128(const float* g, float* l) {
#if __has_builtin(__builtin_amdgcn_global_load_async_to_lds_b128)
  __builtin_amdgcn_global_load_async_to_lds_b128(
      (AS1 v4i*)g, (AS3 v4i*)l, /*imm offset*/0, /*cpol*/0);
#else
  unsigned loff = (unsigned)(unsigned long)(AS3 float*)l;
  asm volatile("global_load_async_to_lds_b128 %0, %1, off"
               :: "v"(loff), "v"(g) : "memory");
#endif
}

__device__ __forceinline__ void wait_async0() {
#if __has_builtin(__builtin_amdgcn_s_wait_asynccnt)
  __builtin_amdgcn_s_wait_asynccnt(0);
#else
  asm volatile("s_wait_asynccnt 0" ::: "memory");
#endif
}

__device__ __forceinline__ float waveReduceSum(float v) {
  v += __shfl_xor(v, 16, 32);
  v += __shfl_xor(v,  8, 32);
  v += __shfl_xor(v,  4, 32);
  v += __shfl_xor(v,  2, 32);
  v += __shfl_xor(v,  1, 32);
  return v;
}

// Broadcast lane `i`'s value to all lanes as a wave-uniform scalar.
// `i` is a compile-time constant under full unroll -> v_readlane_b32.
__device__ __forceinline__ float laneBcast(float v, int i) {
  return __builtin_bit_cast(float,
           __builtin_amdgcn_readlane(__builtin_bit_cast(int, v), i));
}

// Stage 1: one wave32 per batch row. Right-looking forward substitution on
// L built from exp(packed tril), softplus on exp'd diagonal.
// Lane k owns row k of L (its packed segment starts at 32 + k(k+1)/2).
__global__ __launch_bounds__(256) void mvn_nll_stage1(
    const float* __restrict__ mean, const float* __restrict__ y_true,
    float* __restrict__ partials, int B) {
  constexpr int D = 32;
  constexpr int ROWLEN = D + (D * (D + 1)) / 2;  // 560 floats = 2240 B
  __shared__ __align__(16) float smem[8 * ROWLEN];
  __shared__ float wsum[8];

  const int lane = threadIdx.x & 31;
  const int wave = threadIdx.x >> 5;
  const int row  = blockIdx.x * 8 + wave;

  float t = 0.0f;
  if (row < B) {
    const float* grow = mean + (size_t)row * ROWLEN;
    float* rowlds = smem + wave * ROWLEN;

    // Stage entire row into LDS: 4 full B128 transfers + masked tail.
#pragma unroll
    for (int c = 0; c < 4; ++c)
      async_g2l_b128(grow + c * 128 + lane * 4, rowlds + c * 128 + lane * 4);
    if (lane < 12)
      async_g2l_b128(grow + 512 + lane * 4, rowlds + 512 + lane * 4);

    // Overlap: fetch y while the async DMA is in flight.
    float yv = y_true[(size_t)row * D + lane];

    wait_async0();

    const int base = D + (lane * (lane + 1)) / 2;  // lane k's packed L-row

    // Parallel pre-pass (off the serial chain): lane k's diagonal terms.
    // bank(base+k) = k(k+3)/2 mod 64: distinct for all lanes -> no conflicts.
    float diff  = yv - rowlds[lane];               // pred = first 32 floats
    float dval  = log1pf(expf(expf(rowlds[base + lane])));  // softplus(exp(.))
    float rdval = 1.0f / dval;
    float logd  = logf(dval);

    // Serial solve: z_i = (diff_i - acc_i) / dval_i; acc_k += L_ki * z_i.
    // Per step: 1 conflict-free ds_load + exp (off-chain) + readlane + fma.
    float acc = 0.0f;
    float z   = 0.0f;
#pragma unroll
    for (int i = 0; i < D; ++i) {
      float a = expf(rowlds[base + i]);        // L_{lane,i}; garbage if i>lane
      z = (lane == i) ? (diff - acc) * rdval : z;
      float zi = laneBcast(z, i);              // wave-uniform z_i (SGPR)
      acc = (lane > i) ? fmaf(a, zi, acc) : acc;  // cndmask kills bad lanes
    }
    t = fmaf(0.5f * z, z, logd);  // per-lane: 0.5*z_k^2 + log(diag_k)
  }

  t = waveReduceSum(t);
  if (lane == 0) wsum[wave] = t;
  __syncthreads();
  if (threadIdx.x == 0) {
    float bs = 0.0f;
#pragma unroll
    for (int w = 0; w < 8; ++w) bs += wsum[w];
    partials[blockIdx.x] = bs;
  }
}

// Stage 2: reduce per-block partials in f64, add constant, write scalar.
__global__ __launch_bounds__(256) void mvn_nll_stage2(
    const float* __restrict__ partials, int n, float* __restrict__ out,
    double invB) {
  __shared__ double sh[256];
  double acc = 0.0;
  for (int i = threadIdx.x; i < n; i += 256) acc += (double)partials[i];
  sh[threadIdx.x] = acc;
  __syncthreads();
  for (int s = 128; s > 0; s >>= 1) {
    if ((int)threadIdx.x < s) sh[threadIdx.x] += sh[threadIdx.x + s];
    __syncthreads();
  }
  if (threadIdx.x == 0) {
    // + 0.5 * D * log(2*pi), D = 32
    double v = sh[0] * invB + 16.0 * 1.8378770664093454;
    out[0] = (float)v;
  }
}

extern "C" void kernel_launch(void* const* d_in, const int* in_sizes, int n_in,
                              void* d_out, int out_size, void* d_ws, size_t ws_size,
                              hipStream_t stream) {
  const float* mean   = (const float*)d_in[0];
  const float* y_true = (const float*)d_in[1];
  const int B = in_sizes[1] / 32;          // 131072
  const int nblocks = (B + 7) / 8;         // 8 rows (waves) per block
  float* partials = (float*)d_ws;          // nblocks floats (64 KB)

  mvn_nll_stage1<<<nblocks, 256, 0, stream>>>(mean, y_true, partials, B);
  mvn_nll_stage2<<<1, 256, 0, stream>>>(partials, nblocks, (float*)d_out,
                                        1.0 / (double)B);
}